// FastICABaseline_2731599201023
// MI455X (gfx1250) — compile-verified
//
#include <hip/hip_runtime.h>
#include <math.h>

typedef __attribute__((ext_vector_type(2))) float v2f;
typedef __attribute__((ext_vector_type(8))) float v8f;

#define B_      64
#define C_      16
#define T_      32768
#define N_      4
#define MAXIT_  200

#define GRAM_SPLITS 16
#define NB_ITER     8
#define CHUNK_ITER  (T_ / NB_ITER)   // 4096
#define NBS         (T_ / 256)       // 128

// ---------- workspace layout (float offsets) ----------
#define WS_MEAN    0                        // B_*C_            = 1024
#define WS_G       1024                     // B_*256           = 16384
#define WS_K       17408                    // B_*64            = 4096
#define WS_W       21504                    // B_*16            = 1024
#define WS_PART    22528                    // B_*NB_ITER*20    = 10240
#define WS_SCALE   32768                    // B_*4             = 256
#define WS_SUMP    33024                    // B_*NBS*4         = 32768
#define WS_SUMSQP  65792                    // B_*NBS*4         = 32768
#define WS_X1      98560                    // B_*4*T_          = 8388608

__device__ __forceinline__ float waveReduceSum(float v) {
  #pragma unroll
  for (int off = 16; off > 0; off >>= 1) v += __shfl_down(v, off, 32);
  return v;
}

// ---------------- mean over T per (b,c) ----------------
__global__ void fica_mean_kernel(const float* __restrict__ x, float* __restrict__ mean) {
  int bc = blockIdx.x;
  const float* row = x + (size_t)bc * T_;
  float s = 0.f;
  for (int t = threadIdx.x; t < T_; t += blockDim.x) s += row[t];
  __shared__ float lds[8];
  int lane = threadIdx.x & 31, wave = threadIdx.x >> 5;
  s = waveReduceSum(s);
  if (lane == 0) lds[wave] = s;
  __syncthreads();
  if (threadIdx.x == 0) {
    float tot = 0.f;
    for (int w = 0; w < 8; ++w) tot += lds[w];
    mean[bc] = tot / (float)T_;
  }
}

__global__ void fica_zero_kernel(float* __restrict__ p, int n) {
  int i = blockIdx.x * blockDim.x + threadIdx.x;
  if (i < n) p[i] = 0.f;
}

// ---------------- Gram: G[b] += Xc(16x4) x Xc^T(4x16) via WMMA f32 ----------------
// A layout (16x4 f32): lanes 0-15 -> M=lane, {K0,K1}; lanes 16-31 -> M=lane-16, {K2,K3}
// B layout (4x16 f32): lanes 0-15 -> N=lane, {K0,K1}; lanes 16-31 -> N=lane-16, {K2,K3}
// For the Gram both operands are identical per-lane loads of Xc[r][kb+koff..+1].
__global__ void fica_gram_kernel(const float* __restrict__ x,
                                 const float* __restrict__ mean,
                                 float* __restrict__ G) {
  int b    = blockIdx.y;
  int lane = threadIdx.x;            // 32 threads = 1 wave, EXEC all ones
  int r    = lane & 15;
  int koff = (lane >> 4) << 1;       // 0 or 2
  const float* row = x + ((size_t)b * C_ + r) * T_;
  float mr = mean[b * C_ + r];
  v8f acc = {0.f, 0.f, 0.f, 0.f, 0.f, 0.f, 0.f, 0.f};
  int t0 = blockIdx.x * (T_ / GRAM_SPLITS);
  int t1 = t0 + (T_ / GRAM_SPLITS);
  for (int t = t0; t < t1; t += 4) {
    const float* p = row + t + koff;   // 8-byte aligned
    float a0 = p[0] - mr;
    float a1 = p[1] - mr;
    v2f a; a.x = a0; a.y = a1;
    acc = __builtin_amdgcn_wmma_f32_16x16x4_f32(
        /*neg_a=*/false, a, /*neg_b=*/false, a,
        /*c_mod=*/(short)0, acc, /*reuse_a=*/false, /*reuse_b=*/false);
  }
  // C/D layout: VGPR v, lanes 0-15: M=v, N=lane; lanes 16-31: M=v+8, N=lane-16
  int mbase = (lane >> 4) << 3;
  float* Gb = G + (size_t)b * 256;
  #pragma unroll
  for (int v = 0; v < 8; ++v)
    atomicAdd(&Gb[(mbase + v) * 16 + r], acc[v]);
}

// ---------------- 16x16 eigh (Jacobi) + whitening K ----------------
__global__ void fica_eigh16_kernel(const float* __restrict__ G, float* __restrict__ K) {
  int b = blockIdx.x;
  __shared__ float A[256];
  __shared__ float V[256];
  for (int i = threadIdx.x; i < 256; i += blockDim.x) {
    A[i] = G[(size_t)b * 256 + i];
    V[i] = ((i >> 4) == (i & 15)) ? 1.f : 0.f;
  }
  __syncthreads();
  if (threadIdx.x == 0) {
    const int n = 16;
    for (int sweep = 0; sweep < 12; ++sweep)
      for (int p = 0; p < n - 1; ++p)
        for (int q = p + 1; q < n; ++q) {
          float apq = A[p * n + q];
          if (fabsf(apq) < 1e-12f) continue;
          float app = A[p * n + p], aqq = A[q * n + q];
          float tau = (aqq - app) / (2.f * apq);
          float t = (tau >= 0.f ? 1.f : -1.f) / (fabsf(tau) + sqrtf(1.f + tau * tau));
          float c = 1.f / sqrtf(1.f + t * t);
          float s = t * c;
          for (int k = 0; k < n; ++k) {
            float akp = A[k * n + p], akq = A[k * n + q];
            A[k * n + p] = c * akp - s * akq;
            A[k * n + q] = s * akp + c * akq;
          }
          for (int k = 0; k < n; ++k) {
            float apk = A[p * n + k], aqk = A[q * n + k];
            A[p * n + k] = c * apk - s * aqk;
            A[q * n + k] = s * apk + c * aqk;
          }
          for (int k = 0; k < n; ++k) {
            float vkp = V[k * n + p], vkq = V[k * n + q];
            V[k * n + p] = c * vkp - s * vkq;
            V[k * n + q] = s * vkp + c * vkq;
          }
        }
    // top-4 eigenvalues, descending
    int idx[16]; float ev[16];
    for (int i = 0; i < 16; ++i) { idx[i] = i; ev[i] = A[i * 16 + i]; }
    for (int i = 0; i < 4; ++i) {
      int best = i;
      for (int j = i + 1; j < 16; ++j) if (ev[idx[j]] > ev[idx[best]]) best = j;
      int tmp = idx[i]; idx[i] = idx[best]; idx[best] = tmp;
    }
    for (int i = 0; i < 4; ++i) {
      float d = sqrtf(fmaxf(ev[idx[i]], 1e-20f));
      for (int c = 0; c < 16; ++c)
        K[((size_t)b * 4 + i) * 16 + c] = V[c * 16 + idx[i]] / d;
    }
  }
}

// ---------------- X1 = sqrt(T) * K @ (x - mean) ----------------
__global__ void fica_x1_kernel(const float* __restrict__ x, const float* __restrict__ mean,
                               const float* __restrict__ K, float* __restrict__ X1) {
  int b = blockIdx.y;
  int t = blockIdx.x * blockDim.x + threadIdx.x;
  __shared__ float Ks[64];
  __shared__ float off[4];
  if (threadIdx.x < 64) Ks[threadIdx.x] = K[(size_t)b * 64 + threadIdx.x];
  __syncthreads();
  if (threadIdx.x < 4) {
    float o = 0.f;
    for (int c = 0; c < 16; ++c) o -= Ks[threadIdx.x * 16 + c] * mean[b * 16 + c];
    off[threadIdx.x] = o;
  }
  __syncthreads();
  float a0 = off[0], a1 = off[1], a2 = off[2], a3 = off[3];
  #pragma unroll
  for (int c = 0; c < 16; ++c) {
    float xv = x[((size_t)b * C_ + c) * T_ + t];
    a0 += Ks[0 * 16 + c] * xv;
    a1 += Ks[1 * 16 + c] * xv;
    a2 += Ks[2 * 16 + c] * xv;
    a3 += Ks[3 * 16 + c] * xv;
  }
  const float sq = 181.019336f;   // sqrt(32768)
  float* X1b = X1 + (size_t)b * 4 * T_;
  X1b[0 * T_ + t] = sq * a0;
  X1b[1 * T_ + t] = sq * a1;
  X1b[2 * T_ + t] = sq * a2;
  X1b[3 * T_ + t] = sq * a3;
}

// ---------------- 4x4 symmetric decorrelation ----------------
__device__ void fica_jacobi4(float* A, float* V) {
  for (int i = 0; i < 16; ++i) V[i] = 0.f;
  V[0] = V[5] = V[10] = V[15] = 1.f;
  for (int sweep = 0; sweep < 10; ++sweep)
    for (int p = 0; p < 3; ++p)
      for (int q = p + 1; q < 4; ++q) {
        float apq = A[p * 4 + q];
        if (fabsf(apq) < 1e-12f) continue;
        float app = A[p * 4 + p], aqq = A[q * 4 + q];
        float tau = (aqq - app) / (2.f * apq);
        float t = (tau >= 0.f ? 1.f : -1.f) / (fabsf(tau) + sqrtf(1.f + tau * tau));
        float c = 1.f / sqrtf(1.f + t * t);
        float s = t * c;
        for (int k = 0; k < 4; ++k) {
          float akp = A[k * 4 + p], akq = A[k * 4 + q];
          A[k * 4 + p] = c * akp - s * akq;
          A[k * 4 + q] = s * akp + c * akq;
        }
        for (int k = 0; k < 4; ++k) {
          float apk = A[p * 4 + k], aqk = A[q * 4 + k];
          A[p * 4 + k] = c * apk - s * aqk;
          A[q * 4 + k] = s * apk + c * aqk;
        }
        for (int k = 0; k < 4; ++k) {
          float vkp = V[k * 4 + p], vkq = V[k * 4 + q];
          V[k * 4 + p] = c * vkp - s * vkq;
          V[k * 4 + q] = s * vkp + c * vkq;
        }
      }
}

__device__ void fica_sym_decorr4(float* W) {
  float M[16], V[16];
  for (int i = 0; i < 4; ++i)
    for (int j = 0; j < 4; ++j) {
      float s = 0.f;
      for (int k = 0; k < 4; ++k) s += W[i * 4 + k] * W[j * 4 + k];
      M[i * 4 + j] = s;
    }
  fica_jacobi4(M, V);
  float is[4];
  for (int k = 0; k < 4; ++k) is[k] = rsqrtf(fmaxf(M[k * 4 + k], 1e-20f));
  float P[16];
  for (int i = 0; i < 4; ++i)
    for (int j = 0; j < 4; ++j) {
      float s = 0.f;
      for (int k = 0; k < 4; ++k) s += V[i * 4 + k] * is[k] * V[j * 4 + k];
      P[i * 4 + j] = s;
    }
  float Wn[16];
  for (int i = 0; i < 4; ++i)
    for (int j = 0; j < 4; ++j) {
      float s = 0.f;
      for (int k = 0; k < 4; ++k) s += P[i * 4 + k] * W[k * 4 + j];
      Wn[i * 4 + j] = s;
    }
  for (int i = 0; i < 16; ++i) W[i] = Wn[i];
}

__global__ void fica_w0_kernel(const float* __restrict__ w_init, float* __restrict__ W) {
  int b = blockIdx.x * blockDim.x + threadIdx.x;
  if (b >= B_) return;
  float Wl[16];
  for (int i = 0; i < 16; ++i) Wl[i] = w_init[i];
  fica_sym_decorr4(Wl);
  for (int i = 0; i < 16; ++i) W[(size_t)b * 16 + i] = Wl[i];
}

// ---------------- iteration: partial reductions of g@X1^T and sum(Y^2) ----------------
__global__ void fica_iterA_kernel(const float* __restrict__ X1, const float* __restrict__ W,
                                  float* __restrict__ partial,
                                  const int* __restrict__ max_iter, int iter) {
  if (iter >= max_iter[0]) return;
  int b = blockIdx.y;
  int blk = blockIdx.x;
  __shared__ float Ws[16];
  if (threadIdx.x < 16) Ws[threadIdx.x] = W[(size_t)b * 16 + threadIdx.x];
  __syncthreads();
  float accG[16] = {0.f}; float accD[4] = {0.f};
  const float* X1b = X1 + (size_t)b * 4 * T_;
  int t0 = blk * CHUNK_ITER;
  for (int t = t0 + threadIdx.x; t < t0 + CHUNK_ITER; t += blockDim.x) {
    float x0 = X1b[0 * T_ + t], x1v = X1b[1 * T_ + t];
    float x2 = X1b[2 * T_ + t], x3 = X1b[3 * T_ + t];
    float y[4];
    #pragma unroll
    for (int n = 0; n < 4; ++n)
      y[n] = Ws[n * 4 + 0] * x0 + Ws[n * 4 + 1] * x1v + Ws[n * 4 + 2] * x2 + Ws[n * 4 + 3] * x3;
    #pragma unroll
    for (int n = 0; n < 4; ++n) {
      float y2 = y[n] * y[n];
      float g = y2 * y[n];
      accD[n] += y2;
      accG[n * 4 + 0] += g * x0;
      accG[n * 4 + 1] += g * x1v;
      accG[n * 4 + 2] += g * x2;
      accG[n * 4 + 3] += g * x3;
    }
  }
  __shared__ float lds[8][20];
  int lane = threadIdx.x & 31, wave = threadIdx.x >> 5;
  #pragma unroll
  for (int j = 0; j < 16; ++j) { float v = waveReduceSum(accG[j]); if (lane == 0) lds[wave][j] = v; }
  #pragma unroll
  for (int j = 0; j < 4; ++j) { float v = waveReduceSum(accD[j]); if (lane == 0) lds[wave][16 + j] = v; }
  __syncthreads();
  if (threadIdx.x < 20) {
    float s = 0.f;
    for (int w = 0; w < 8; ++w) s += lds[w][threadIdx.x];
    partial[((size_t)b * NB_ITER + blk) * 20 + threadIdx.x] = s;
  }
}

__global__ void fica_iterB_kernel(float* __restrict__ W, const float* __restrict__ partial,
                                  const int* __restrict__ max_iter, int iter) {
  if (iter >= max_iter[0]) return;
  int b = blockIdx.x * blockDim.x + threadIdx.x;
  if (b >= B_) return;
  float sum[20];
  for (int j = 0; j < 20; ++j) {
    float s = 0.f;
    for (int blk = 0; blk < NB_ITER; ++blk)
      s += partial[((size_t)b * NB_ITER + blk) * 20 + j];
    sum[j] = s;
  }
  float Wl[16];
  for (int i = 0; i < 16; ++i) Wl[i] = W[(size_t)b * 16 + i];
  const float invT = 1.f / (float)T_;
  float Wn[16];
  for (int n = 0; n < 4; ++n) {
    float gder = 3.f * sum[16 + n] * invT;
    for (int m = 0; m < 4; ++m)
      Wn[n * 4 + m] = sum[n * 4 + m] * invT - gder * Wl[n * 4 + m];
  }
  fica_sym_decorr4(Wn);
  for (int i = 0; i < 16; ++i) W[(size_t)b * 16 + i] = Wn[i];
}

// ---------------- final: S = W@X1, then normalize by std ----------------
__global__ void fica_s1_kernel(const float* __restrict__ X1, const float* __restrict__ W,
                               float* __restrict__ out, float* __restrict__ sumP,
                               float* __restrict__ sumsqP) {
  int b = blockIdx.y;
  __shared__ float Ws[16];
  if (threadIdx.x < 16) Ws[threadIdx.x] = W[(size_t)b * 16 + threadIdx.x];
  __syncthreads();
  int t = blockIdx.x * blockDim.x + threadIdx.x;
  const float* X1b = X1 + (size_t)b * 4 * T_;
  float x0 = X1b[t], x1v = X1b[T_ + t], x2 = X1b[2 * T_ + t], x3 = X1b[3 * T_ + t];
  float y[4], ysq[4];
  #pragma unroll
  for (int n = 0; n < 4; ++n) {
    y[n] = Ws[n * 4 + 0] * x0 + Ws[n * 4 + 1] * x1v + Ws[n * 4 + 2] * x2 + Ws[n * 4 + 3] * x3;
    out[((size_t)b * 4 + n) * T_ + t] = y[n];
    ysq[n] = y[n] * y[n];
  }
  __shared__ float lds[8][8];
  int lane = threadIdx.x & 31, wave = threadIdx.x >> 5;
  #pragma unroll
  for (int n = 0; n < 4; ++n) {
    float v = waveReduceSum(y[n]);   if (lane == 0) lds[wave][n] = v;
    float w = waveReduceSum(ysq[n]); if (lane == 0) lds[wave][4 + n] = w;
  }
  __syncthreads();
  if (threadIdx.x < 8) {
    float s = 0.f;
    for (int w = 0; w < 8; ++w) s += lds[w][threadIdx.x];
    if (threadIdx.x < 4)
      sumP[((size_t)b * NBS + blockIdx.x) * 4 + threadIdx.x] = s;
    else
      sumsqP[((size_t)b * NBS + blockIdx.x) * 4 + (threadIdx.x - 4)] = s;
  }
}

__global__ void fica_s2_kernel(const float* __restrict__ sumP, const float* __restrict__ sumsqP,
                               float* __restrict__ scale) {
  int i = blockIdx.x * blockDim.x + threadIdx.x;
  if (i >= B_ * 4) return;
  int b = i >> 2, n = i & 3;
  float s = 0.f, sq = 0.f;
  for (int blk = 0; blk < NBS; ++blk) {
    s  += sumP[((size_t)b * NBS + blk) * 4 + n];
    sq += sumsqP[((size_t)b * NBS + blk) * 4 + n];
  }
  float m = s / (float)T_;
  float var = sq / (float)T_ - m * m;
  scale[i] = rsqrtf(fmaxf(var, 1e-20f));
}

__global__ void fica_s3_kernel(float* __restrict__ out, const float* __restrict__ scale) {
  int b = blockIdx.y;
  int t = blockIdx.x * blockDim.x + threadIdx.x;
  #pragma unroll
  for (int n = 0; n < 4; ++n)
    out[((size_t)b * 4 + n) * T_ + t] *= scale[b * 4 + n];
}

extern "C" void kernel_launch(void* const* d_in, const int* in_sizes, int n_in,
                              void* d_out, int out_size, void* d_ws, size_t ws_size,
                              hipStream_t stream) {
  (void)in_sizes; (void)n_in; (void)out_size; (void)ws_size;
  const float* x      = (const float*)d_in[0];
  const float* w_init = (const float*)d_in[1];
  const int*   mi     = (const int*)d_in[2];
  float* out = (float*)d_out;
  float* ws  = (float*)d_ws;

  float* mean   = ws + WS_MEAN;
  float* G      = ws + WS_G;
  float* K      = ws + WS_K;
  float* W      = ws + WS_W;
  float* part   = ws + WS_PART;
  float* scale  = ws + WS_SCALE;
  float* sumP   = ws + WS_SUMP;
  float* sumsqP = ws + WS_SUMSQP;
  float* X1     = ws + WS_X1;

  fica_mean_kernel<<<B_ * C_, 256, 0, stream>>>(x, mean);
  fica_zero_kernel<<<(B_ * 256 + 255) / 256, 256, 0, stream>>>(G, B_ * 256);
  fica_gram_kernel<<<dim3(GRAM_SPLITS, B_), 32, 0, stream>>>(x, mean, G);
  fica_eigh16_kernel<<<B_, 32, 0, stream>>>(G, K);
  fica_x1_kernel<<<dim3(T_ / 256, B_), 256, 0, stream>>>(x, mean, K, X1);
  fica_w0_kernel<<<1, 64, 0, stream>>>(w_init, W);
  for (int it = 0; it < MAXIT_; ++it) {
    fica_iterA_kernel<<<dim3(NB_ITER, B_), 256, 0, stream>>>(X1, W, part, mi, it);
    fica_iterB_kernel<<<1, 64, 0, stream>>>(W, part, mi, it);
  }
  fica_s1_kernel<<<dim3(NBS, B_), 256, 0, stream>>>(X1, W, out, sumP, sumsqP);
  fica_s2_kernel<<<1, 256, 0, stream>>>(sumP, sumsqP, scale);
  fica_s3_kernel<<<dim3(NBS, B_), 256, 0, stream>>>(out, scale);
}